// MultiHeadMixer_1486058684867
// MI455X (gfx1250) — compile-verified
//
#include <hip/hip_runtime.h>
#include <stdint.h>

// MI455X / gfx1250, wave32. FP32 WMMA path: V_WMMA_F32_16X16X4_F32.
typedef __attribute__((ext_vector_type(2))) float v2f;  // A/B frag: 16x4 f32 = 2 VGPRs
typedef __attribute__((ext_vector_type(8))) float v8f;  // C/D frag: 16x16 f32 = 8 VGPRs

#define B_ 4
#define S_ 2048
#define E_ 1024
#define H_ 16
#define D_ 64

static __device__ __forceinline__ v8f wmma_f32(v2f a, v2f b, v8f c) {
  return __builtin_amdgcn_wmma_f32_16x16x4_f32(
      /*neg_a=*/false, a, /*neg_b=*/false, b,
      /*c_mod=*/(short)0, c, /*reuse_a=*/false, /*reuse_b=*/false);
}

// Async copy global -> LDS (CDNA5 TDM-adjacent path, tracked by ASYNCcnt).
// vdst operand carries the LDS byte address; vaddr the 64-bit global address.
static __device__ __forceinline__ void async_g2l_b128(uint32_t lds_byte_addr,
                                                      const void* gaddr) {
  asm volatile("global_load_async_to_lds_b128 %0, %1, off"
               :: "v"(lds_byte_addr), "v"(gaddr)
               : "memory");
}
static __device__ __forceinline__ void wait_async_le4() {
  asm volatile("s_wait_asynccnt 0x4" ::: "memory");
}
static __device__ __forceinline__ void wait_async_le0() {
  asm volatile("s_wait_asynccnt 0x0" ::: "memory");
}

// -------------------------------------------------------------------------
// Per-head inclusive cumsum of weight[h, :] -> norm[h, :]
// -------------------------------------------------------------------------
__global__ __launch_bounds__(256) void head_cumsum(const float* __restrict__ w,
                                                   float* __restrict__ norm) {
  __shared__ float part[256];
  const int h = blockIdx.x;
  const int tid = threadIdx.x;
  const float* wr = w + (size_t)h * S_;
  float v[8];
  float run = 0.0f;
#pragma unroll
  for (int i = 0; i < 8; ++i) { run += wr[tid * 8 + i]; v[i] = run; }
  part[tid] = run;
  __syncthreads();
  if (tid == 0) {
    float acc = 0.0f;
    for (int i = 0; i < 256; ++i) { float t = part[i]; part[i] = acc; acc += t; }
  }
  __syncthreads();
  const float off = part[tid];
  float* nr = norm + (size_t)h * S_;
#pragma unroll
  for (int i = 0; i < 8; ++i) nr[tid * 8 + i] = v[i] + off;
}

// -------------------------------------------------------------------------
// C[M,N] = A[M,K] @ Bw[N,K]^T (+ bias[N]).  A, Bw row-major, K contiguous.
// Block: 256 thr = 8 waves; block tile 128x128; wave tile 32(M) x 64(N).
// K-stage = 16, double-buffered LDS filled with global_load_async_to_lds_b128
// (each wave: 4 b128 loads per stage, each covering 8 full 64B rows).  Fragments read
// from LDS via conflict-free ds_load_b64 (row stride 20 floats: 16B-aligned
// rows, r*20 mod 64 distinct over r=0..15, hk-half uses banks ≡2 mod 4).
// -------------------------------------------------------------------------
#define BK 16
#define LDSW 20                    // padded row stride in floats
#define ABUF (128 * LDSW)          // floats per tile buffer

__global__ __launch_bounds__(256) void gemm_nt_f32(const float* __restrict__ A,
                                                   const float* __restrict__ Bw,
                                                   const float* __restrict__ bias,
                                                   float* __restrict__ C,
                                                   int M, int N, int K) {
  __shared__ float smem[4 * ABUF];  // A0 | A1 | B0 | B1  (40 KB)

  const int tid  = threadIdx.x;
  const int lane = tid & 31;
  const int wave = tid >> 5;
  const int r  = lane & 15;   // row (A) / col (B) within 16
  const int hk = lane >> 4;   // K half-select

  const int bm = blockIdx.x * 128;
  const int bn = blockIdx.y * 128;
  const int mw = (wave >> 1) * 32;   // wave M offset in block tile
  const int nw = (wave & 1) * 64;    // wave N offset in block tile

  const uint32_t ldsbase = (uint32_t)(uintptr_t)smem;
  const int frow = tid >> 2;        // fill row 0..63 (+64 with pass)
  const int fseg = tid & 3;         // 16B segment within a 64B row-slab
  const float* gA = A  + (size_t)bm * K;
  const float* gB = Bw + (size_t)bn * K;

  auto fill = [&](int buf, int k0) {
#pragma unroll
    for (int p = 0; p < 2; ++p) {
      const int row = frow + p * 64;
      const uint32_t la = ldsbase +
          (uint32_t)(((buf * ABUF) + row * LDSW) * 4 + fseg * 16);
      const uint32_t lb = la + (uint32_t)(2 * ABUF * 4);
      async_g2l_b128(la, gA + (size_t)row * K + k0 + fseg * 4);
      async_g2l_b128(lb, gB + (size_t)row * K + k0 + fseg * 4);
    }
  };

  v8f acc[2][4] = {};
  const int nst = K / BK;

  fill(0, 0);
  for (int s = 0; s < nst; ++s) {
    const int cur = s & 1;
    if (s + 1 < nst) {
      fill(cur ^ 1, (s + 1) * BK);
      wait_async_le4();   // in-order completion: current stage's 4 are done
    } else {
      wait_async_le0();
    }
    __syncthreads();

    const float* Ac = smem + cur * ABUF;
    const float* Bc = smem + (2 + cur) * ABUF;
#pragma unroll
    for (int kk = 0; kk < BK; kk += 4) {
      const int ko = kk + hk * 2;
      const v2f a0 = *(const v2f*)&Ac[(mw +  0 + r) * LDSW + ko];
      const v2f a1 = *(const v2f*)&Ac[(mw + 16 + r) * LDSW + ko];
      const v2f b0 = *(const v2f*)&Bc[(nw +  0 + r) * LDSW + ko];
      const v2f b1 = *(const v2f*)&Bc[(nw + 16 + r) * LDSW + ko];
      const v2f b2 = *(const v2f*)&Bc[(nw + 32 + r) * LDSW + ko];
      const v2f b3 = *(const v2f*)&Bc[(nw + 48 + r) * LDSW + ko];
      acc[0][0] = wmma_f32(a0, b0, acc[0][0]);
      acc[0][1] = wmma_f32(a0, b1, acc[0][1]);
      acc[0][2] = wmma_f32(a0, b2, acc[0][2]);
      acc[0][3] = wmma_f32(a0, b3, acc[0][3]);
      acc[1][0] = wmma_f32(a1, b0, acc[1][0]);
      acc[1][1] = wmma_f32(a1, b1, acc[1][1]);
      acc[1][2] = wmma_f32(a1, b2, acc[1][2]);
      acc[1][3] = wmma_f32(a1, b3, acc[1][3]);
    }
    __syncthreads();
  }

  // C/D layout: VGPR e -> M = e (lanes 0-15) / 8+e (lanes 16-31); N = lane%16
#pragma unroll
  for (int i = 0; i < 2; ++i) {
#pragma unroll
    for (int j = 0; j < 4; ++j) {
      const int nn = bn + nw + j * 16 + r;
      const float bv = bias ? bias[nn] : 0.0f;
#pragma unroll
      for (int e = 0; e < 8; ++e) {
        const int mm = bm + mw + i * 16 + hk * 8 + e;
        C[(size_t)mm * N + nn] = acc[i][j][e] + bv;
      }
    }
  }
}

// -------------------------------------------------------------------------
// Causal Toeplitz mix per (b,h):
//   Y2[b,t,h*64+d] = ( sum_{s<=t} Y1[b,s,h*64+d] * w[h,t-s] ) / norm[h,t]
//                    + bias[h,t]
// One block per (b*h, t-chunk of 128). 8 waves, each 16(t) x 64(d).
// Toeplitz A-fragment built from the LDS-staged weight row; triangular
// structure skips s-blocks above the diagonal entirely.
// -------------------------------------------------------------------------
__global__ __launch_bounds__(256) void toeplitz_mix(const float* __restrict__ Y1,
                                                    const float* __restrict__ weight,
                                                    const float* __restrict__ biasHS,
                                                    const float* __restrict__ norm,
                                                    float* __restrict__ Y2) {
  __shared__ float wl[S_];  // 8 KB

  const int bh = blockIdx.x;  // 0..B*H-1
  const int b = bh / H_;
  const int h = bh % H_;
  const int lane = threadIdx.x & 31;
  const int wave = threadIdx.x >> 5;
  const int r  = lane & 15;
  const int hk = lane >> 4;

  for (int i = threadIdx.x; i < S_; i += 256) wl[i] = weight[(size_t)h * S_ + i];
  __syncthreads();

  const int t0 = blockIdx.y * 128 + wave * 16;
  const int m  = t0 + r;  // this lane's t-row in the A fragment
  const float* Xp = Y1 + (size_t)b * S_ * E_ + h * D_;  // X[s][d] = Xp[s*E_+d]

  v8f acc[4] = {};

  for (int sb = 0; sb <= t0; sb += 16) {
#pragma unroll
    for (int kk = 0; kk < 16; kk += 4) {
      const int s0 = sb + kk + hk * 2;
      const int off0 = m - s0;
      const int off1 = off0 - 1;
      v2f av;
      av[0] = (off0 >= 0) ? wl[off0 >= 0 ? off0 : 0] : 0.0f;
      av[1] = (off1 >= 0) ? wl[off1 >= 0 ? off1 : 0] : 0.0f;
      const float* X0 = Xp + (size_t)s0 * E_;
#pragma unroll
      for (int j = 0; j < 4; ++j) {
        const int col = j * 16 + r;
        v2f bv;
        bv[0] = X0[col];        // row s0
        bv[1] = X0[E_ + col];   // row s0+1
        acc[j] = wmma_f32(av, bv, acc[j]);
      }
    }
  }

  float* Op = Y2 + (size_t)b * S_ * E_ + h * D_;
#pragma unroll
  for (int j = 0; j < 4; ++j) {
    const int d = j * 16 + r;
#pragma unroll
    for (int e = 0; e < 8; ++e) {
      const int t = t0 + hk * 8 + e;
      const float nv = norm[(size_t)h * S_ + t];
      const float bb = biasHS[(size_t)h * S_ + t];
      Op[(size_t)t * E_ + d] = acc[j][e] / nv + bb;
    }
  }
}

// -------------------------------------------------------------------------
extern "C" void kernel_launch(void* const* d_in, const int* in_sizes, int n_in,
                              void* d_out, int out_size, void* d_ws, size_t ws_size,
                              hipStream_t stream) {
  const float* x      = (const float*)d_in[0];  // (B,S,E)
  const float* weight = (const float*)d_in[1];  // (H,S)
  const float* bias   = (const float*)d_in[2];  // (H,S)
  const float* inp_w  = (const float*)d_in[3];  // (E,E)
  const float* inp_b  = (const float*)d_in[4];  // (E,)
  const float* out_w  = (const float*)d_in[5];  // (E,E)
  float* out = (float*)d_out;

  // Workspace: Y1 (B,S,E) | Y2 (B,S,E) | norm (H,S)  -> ~67.2 MB
  float* Y1   = (float*)d_ws;
  float* Y2   = Y1 + (size_t)B_ * S_ * E_;
  float* norm = Y2 + (size_t)B_ * S_ * E_;

  head_cumsum<<<H_, 256, 0, stream>>>(weight, norm);

  const int M = B_ * S_;          // 8192
  dim3 gGemm(M / 128, E_ / 128);  // (64, 8)
  gemm_nt_f32<<<gGemm, 256, 0, stream>>>(x, inp_w, inp_b, Y1, M, E_, E_);

  dim3 gMix(B_ * H_, S_ / 128);   // (64, 16)
  toeplitz_mix<<<gMix, 256, 0, stream>>>(Y1, weight, bias, norm, Y2);

  gemm_nt_f32<<<gGemm, 256, 0, stream>>>(Y2, out_w, nullptr, out, M, E_, E_);
}